// FinalLayer_29549374997212
// MI455X (gfx1250) — compile-verified
//
#include <hip/hip_runtime.h>

// ---------------------------------------------------------------------------
// CDNA5 (gfx1250) IRS-MU-MIMO WMMSE FinalLayer.
//  * Big Theta GEMMs: TDM (tensor_load_to_lds) double-buffered LDS staging,
//    4 waves x v_wmma_f32_16x16x4_f32, s_wait_tensorcnt sync.
//  * Small/batched GEMMs: generic strided complex WMMA kernel (full-tile fast
//    path without predication).
//  * HPD solves via batched complex Gauss-Jordan.
// ---------------------------------------------------------------------------

typedef float v2f __attribute__((ext_vector_type(2)));
typedef float v8f __attribute__((ext_vector_type(8)));

#if defined(__has_builtin)
#if __has_builtin(__builtin_amdgcn_tensor_load_to_lds)
#define HAVE_TDM 1
#endif
#endif
#ifndef HAVE_TDM
#define HAVE_TDM 0
#endif

#if HAVE_TDM
typedef unsigned int tdm_u32x4 __attribute__((ext_vector_type(4)));
typedef int          tdm_i32x4 __attribute__((ext_vector_type(4)));
typedef int          tdm_i32x8 __attribute__((ext_vector_type(8)));

// 2D TDM load: rows x rowFloats (fp32), row pitch strideFloats, into LDS.
__device__ __forceinline__ void tdm_load_2d(float* ldsDst, const float* gsrc,
                                            unsigned rows, unsigned rowFloats,
                                            long long strideFloats) {
  const unsigned lds_off = (unsigned)(unsigned long long)ldsDst;  // LDS offset = low 32 bits
  const unsigned long long ga = (unsigned long long)gsrc;
  const unsigned long long st = (unsigned long long)strideFloats;
  tdm_u32x4 g0;
  g0[0] = 1u;                                                 // count=1 (valid user D#)
  g0[1] = lds_off;                                            // lds_addr (bytes)
  g0[2] = (unsigned)ga;                                       // global_addr[31:0]
  g0[3] = (unsigned)((ga >> 32) & 0x01FFFFFFu) | (2u << 30);  // global_addr[56:32] | type=2
  tdm_i32x8 g1;
  g1[0] = (int)(2u << 16);                                    // data_size = 4B
  g1[1] = (int)((rowFloats & 0xFFFFu) << 16);                 // tensor_dim0[15:0]
  g1[2] = (int)((rowFloats >> 16) | ((rows & 0xFFFFu) << 16));// dim0 hi | tensor_dim1 lo
  g1[3] = (int)((rows >> 16) | ((rowFloats & 0xFFFFu) << 16));// dim1 hi | tile_dim0
  g1[4] = (int)(rows & 0xFFFFu);                              // tile_dim1 (tile_dim2=0)
  g1[5] = (int)(st & 0xFFFFFFFFull);                          // tensor_dim0_stride lo
  g1[6] = (int)((st >> 32) & 0xFFFFull);                      // stride hi | dim1_stride lo
  g1[7] = 0;
  tdm_i32x4 gz; gz[0] = 0; gz[1] = 0; gz[2] = 0; gz[3] = 0;
#if __clang_major__ >= 23
  tdm_i32x8 gz8; gz8[0]=0; gz8[1]=0; gz8[2]=0; gz8[3]=0; gz8[4]=0; gz8[5]=0; gz8[6]=0; gz8[7]=0;
  __builtin_amdgcn_tensor_load_to_lds(g0, g1, gz, gz, gz8, 0);
#else
  __builtin_amdgcn_tensor_load_to_lds(g0, g1, gz, gz, 0);
#endif
}

// Stage one complex panel: K==0 planar (re/im planes), K==1 interleaved.
template<int K>
__device__ __forceinline__ void tdm_panel(float* lds, const float* p, long long po,
                                          long long goff, long long strideElems,
                                          int rows, int rowElems) {
  if (K == 0) {
    tdm_load_2d(lds,                   p + goff,      rows, rowElems, strideElems);
    tdm_load_2d(lds + rows * rowElems, p + po + goff, rows, rowElems, strideElems);
  } else {
    tdm_load_2d(lds, p + 2 * goff, rows, 2 * rowElems, 2 * strideElems);
  }
}
#endif  // HAVE_TDM

template<int K>
__device__ __forceinline__ void coop_panel(float* lds, const float* p, long long po,
                                           long long goff, long long strideElems,
                                           int rows, int rowElems, int tid) {
  if (K == 0) {
    for (int e = tid; e < rows * rowElems; e += 128) {
      const int r = e / rowElems, c = e % rowElems;
      lds[e]                   = p[goff + (long long)r * strideElems + c];
      lds[rows * rowElems + e] = p[po + goff + (long long)r * strideElems + c];
    }
  } else {
    for (int e = tid; e < rows * rowElems * 2; e += 128) {
      const int r = e / (2 * rowElems), c = e % (2 * rowElems);
      lds[e] = p[2 * goff + (long long)r * 2 * strideElems + c];
    }
  }
}

struct BigArgs {
  const float* Ap; long long Ao;   // imag plane offset (planar); ignored interleaved
  const float* Bp; long long Bo;
  float* Cp;                        // interleaved complex out
  const float* Dp; long long Do;    // optional planar add source
  int M, N, Kd;
  long long rsA, rsB, rsC, csC, rsD, csD;
  long long sA0, sB0, sC0, sD0;
  int useD;
};

// One 32-deep K chunk from LDS: 8 k-steps x 4 WMMA.
template<int AK, int BK>
__device__ __forceinline__ void big_chunk(const float* sA, const float* sB,
                                          int mr, int half, int ncol,
                                          v8f& cre, v8f& cim) {
#pragma unroll
  for (int k2 = 0; k2 < 32; k2 += 4) {
    v2f ar, ai, br, bi;
#pragma unroll
    for (int t = 0; t < 2; ++t) {
      const int kc = k2 + half * 2 + t;
      if (AK == 0) { ar[t] = sA[mr * 32 + kc];      ai[t] = sA[512 + mr * 32 + kc]; }
      else         { ar[t] = sA[mr * 64 + 2 * kc];  ai[t] = sA[mr * 64 + 2 * kc + 1]; }
      if (BK == 0) { br[t] = sB[kc * 64 + ncol];    bi[t] = sB[2048 + kc * 64 + ncol]; }
      else         { br[t] = sB[kc * 128 + 2 * ncol]; bi[t] = sB[kc * 128 + 2 * ncol + 1]; }
    }
    const v2f nai = -ai;
    cre = __builtin_amdgcn_wmma_f32_16x16x4_f32(false, ar,  false, br, (short)0, cre, false, false);
    cre = __builtin_amdgcn_wmma_f32_16x16x4_f32(false, nai, false, bi, (short)0, cre, false, false);
    cim = __builtin_amdgcn_wmma_f32_16x16x4_f32(false, ar,  false, bi, (short)0, cim, false, false);
    cim = __builtin_amdgcn_wmma_f32_16x16x4_f32(false, ai,  false, br, (short)0, cim, false, false);
  }
}

// Big GEMM: workgroup = 4 waves = 16(M) x 64(N) complex tile; K staged via TDM
// into double-buffered LDS. Requires M%16==0, N%64==0, Kd%32==0, csA=csB=1.
template<int AK, int BK>
__global__ __launch_bounds__(128) void zgemm_big(BigArgs g) {
  __shared__ float sA[2][1024];
  __shared__ float sB[2][4096];
  const int i0 = blockIdx.x * 16, j0 = blockIdx.y * 64;
  const int b = blockIdx.z;
  const int tid = threadIdx.x;
  const int wave = tid >> 5, lane = tid & 31;
  const int half = lane >> 4, mr = lane & 15;
  const int ncol = wave * 16 + mr;
  const long long offA = (long long)b * g.sA0 + (long long)i0 * g.rsA;
  const long long offB = (long long)b * g.sB0 + j0;
  v8f cre = {0.f,0.f,0.f,0.f,0.f,0.f,0.f,0.f};
  v8f cim = {0.f,0.f,0.f,0.f,0.f,0.f,0.f,0.f};
#if HAVE_TDM
  if (wave == 0) {
    tdm_panel<AK>(sA[0], g.Ap, g.Ao, offA, g.rsA, 16, 32);
    tdm_panel<BK>(sB[0], g.Bp, g.Bo, offB, g.rsB, 32, 64);
  }
  for (int kk = 0; kk < g.Kd; kk += 32) {
    const int cur = (kk >> 5) & 1;
    if (wave == 0) __builtin_amdgcn_s_wait_tensorcnt(0);
    __syncthreads();
    if (wave == 0 && kk + 32 < g.Kd) {   // prefetch next chunk while computing
      tdm_panel<AK>(sA[cur ^ 1], g.Ap, g.Ao, offA + kk + 32, g.rsA, 16, 32);
      tdm_panel<BK>(sB[cur ^ 1], g.Bp, g.Bo, offB + (long long)(kk + 32) * g.rsB, g.rsB, 32, 64);
    }
    big_chunk<AK, BK>(sA[cur], sB[cur], mr, half, ncol, cre, cim);
  }
#else
  for (int kk = 0; kk < g.Kd; kk += 32) {
    __syncthreads();
    coop_panel<AK>(sA[0], g.Ap, g.Ao, offA + kk, g.rsA, 16, 32, tid);
    coop_panel<BK>(sB[0], g.Bp, g.Bo, offB + (long long)kk * g.rsB, g.rsB, 32, 64, tid);
    __syncthreads();
    big_chunk<AK, BK>(sA[0], sB[0], mr, half, ncol, cre, cim);
  }
#endif
  const long long offC = (long long)b * g.sC0;
  const long long offD = (long long)b * g.sD0;
#pragma unroll
  for (int i = 0; i < 8; ++i) {
    const int row = i0 + i + half * 8;
    const int col = j0 + ncol;
    float re = cre[i], im = cim[i];
    if (g.useD) {
      const long long id = offD + (long long)row * g.rsD + (long long)col * g.csD;
      re += g.Dp[id]; im += g.Dp[id + g.Do];
    }
    const long long ic = offC + (long long)row * g.rsC + (long long)col * g.csC;
    g.Cp[2 * ic]     = re;
    g.Cp[2 * ic + 1] = im;
  }
}

// Pack planar (2,16,2048,32) -> interleaved (l,d,m) for contiguous TDM rows.
__global__ void packzl_kernel(const float* src, float* dst) {
  const long long idx = (long long)blockIdx.x * 256 + threadIdx.x;  // d*65536 + m*32 + l
  if (idx >= 1048576) return;
  const int d = (int)(idx >> 16);
  const int m = (int)((idx >> 5) & 2047);
  const int l = (int)(idx & 31);
  const long long o = (long long)l * 32768 + (long long)d * 2048 + m;
  dst[2 * o]     = src[idx];
  dst[2 * o + 1] = src[idx + 1048576];
}

// ---------------------------------------------------------------------------
// Generic strided batched complex GEMM (small/medium shapes).
// ---------------------------------------------------------------------------

struct COp { const float* p; long long s; long long o; };

struct GemmArgs {
  COp A, B, D;
  float* Cp; long long Cs, Co;
  int M, N, Kd;                                       // Kd % 4 == 0 required
  long long rsA, csA, rsB, csB, rsC, csC, rsD, csD;
  int nb0, nbatch;
  long long sA0, sA1, sB0, sB1, sC0, sC1, sD0, sD1;
  int nred; long long sAr, sBr;
  float alpha, dscale, diag;
  int conjA, conjB, beta, useD;
};

template<bool CHK>
__device__ __forceinline__ void zgemm_accum(const GemmArgs& g, long long oA, long long oB,
                                            int arow, int bcol, int half,
                                            v8f& cre, v8f& cim) {
  for (int kk = 0; kk < g.Kd; kk += 4) {
    const int kb = kk + half * 2;
    v2f ar, ai, br, bi;
#pragma unroll
    for (int t = 0; t < 2; ++t) {
      const int kc = kb + t;
      float re = 0.f, im = 0.f;
      if (!CHK || arow < g.M) {
        const long long idx = oA + (long long)arow * g.rsA + (long long)kc * g.csA;
        re = g.A.p[idx * g.A.s];
        im = g.A.p[idx * g.A.s + g.A.o];
      }
      ar[t] = re; ai[t] = g.conjA ? -im : im;
      re = 0.f; im = 0.f;
      if (!CHK || bcol < g.N) {
        const long long idx = oB + (long long)kc * g.rsB + (long long)bcol * g.csB;
        re = g.B.p[idx * g.B.s];
        im = g.B.p[idx * g.B.s + g.B.o];
      }
      br[t] = re; bi[t] = g.conjB ? -im : im;
    }
    const v2f nai = -ai;
    cre = __builtin_amdgcn_wmma_f32_16x16x4_f32(false, ar,  false, br, (short)0, cre, false, false);
    cre = __builtin_amdgcn_wmma_f32_16x16x4_f32(false, nai, false, bi, (short)0, cre, false, false);
    cim = __builtin_amdgcn_wmma_f32_16x16x4_f32(false, ar,  false, bi, (short)0, cim, false, false);
    cim = __builtin_amdgcn_wmma_f32_16x16x4_f32(false, ai,  false, br, (short)0, cim, false, false);
  }
}

__global__ __launch_bounds__(32) void zgemm_wmma(GemmArgs g) {
  const int i0 = blockIdx.x * 16, j0 = blockIdx.y * 16;
  const int b  = blockIdx.z;
  const int b0 = b % g.nb0, b1 = b / g.nb0;
  const long long offA = (long long)b0 * g.sA0 + (long long)b1 * g.sA1;
  const long long offB = (long long)b0 * g.sB0 + (long long)b1 * g.sB1;
  const long long offC = (long long)b0 * g.sC0 + (long long)b1 * g.sC1;
  const long long offD = (long long)b0 * g.sD0 + (long long)b1 * g.sD1;
  const int lane = threadIdx.x;
  const int half = lane >> 4;
  const int mr   = lane & 15;
  const int arow = i0 + mr;
  const int bcol = j0 + mr;
  const bool full = (i0 + 16 <= g.M) && (j0 + 16 <= g.N);
  v8f cre = {0.f,0.f,0.f,0.f,0.f,0.f,0.f,0.f};
  v8f cim = {0.f,0.f,0.f,0.f,0.f,0.f,0.f,0.f};
  for (int r = 0; r < g.nred; ++r) {
    const long long oA = offA + (long long)r * g.sAr;
    const long long oB = offB + (long long)r * g.sBr;
    if (full) zgemm_accum<false>(g, oA, oB, arow, bcol, half, cre, cim);
    else      zgemm_accum<true >(g, oA, oB, arow, bcol, half, cre, cim);
  }
#pragma unroll
  for (int i = 0; i < 8; ++i) {
    const int row = i0 + i + half * 8;
    const int col = j0 + mr;
    if (row < g.M && col < g.N) {
      float re = g.alpha * cre[i];
      float im = g.alpha * cim[i];
      if (g.useD) {
        const long long idxD = offD + (long long)row * g.rsD + (long long)col * g.csD;
        re += g.dscale * g.D.p[idxD * g.D.s];
        im += g.dscale * g.D.p[idxD * g.D.s + g.D.o];
      }
      const long long idxC = offC + (long long)row * g.rsC + (long long)col * g.csC;
      if (g.beta) {
        re += g.Cp[idxC * g.Cs];
        im += g.Cp[idxC * g.Cs + g.Co];
      }
      if (row == col) re += g.diag;
      g.Cp[idxC * g.Cs]        = re;
      g.Cp[idxC * g.Cs + g.Co] = im;
    }
  }
}

// Batched complex Gauss-Jordan inverse (HPD). ws per batch: [M | Inv].
__global__ void cinv_kernel(const float* Ap, long long As, long long Ao,
                            long long rsA, long long csA, long long sAb,
                            float* W, long long wstrideF, int n) {
  const int b = blockIdx.x, tid = threadIdx.x, T = blockDim.x;
  float* M = W + (long long)b * wstrideF;
  float* I = M + 2LL * n * n;
  for (int e = tid; e < n * n; e += T) {
    const int r = e / n, c = e % n;
    const long long idx = (long long)b * sAb + (long long)r * rsA + (long long)c * csA;
    M[2*e]   = Ap[idx * As];
    M[2*e+1] = Ap[idx * As + Ao];
    I[2*e]   = (r == c) ? 1.f : 0.f;
    I[2*e+1] = 0.f;
  }
  __syncthreads();
  __shared__ float piv[2];
  for (int p = 0; p < n; ++p) {
    if (tid == 0) {
      const float a = M[2*(p*n+p)], bim = M[2*(p*n+p)+1];
      const float d = a*a + bim*bim + 1e-30f;
      piv[0] = a / d; piv[1] = -bim / d;
    }
    __syncthreads();
    const float pr = piv[0], pi = piv[1];
    for (int c = tid; c < n; c += T) {
      float xr = M[2*(p*n+c)], xi = M[2*(p*n+c)+1];
      M[2*(p*n+c)] = xr*pr - xi*pi; M[2*(p*n+c)+1] = xr*pi + xi*pr;
      xr = I[2*(p*n+c)]; xi = I[2*(p*n+c)+1];
      I[2*(p*n+c)] = xr*pr - xi*pi; I[2*(p*n+c)+1] = xr*pi + xi*pr;
    }
    __syncthreads();
    for (int r = tid; r < n; r += T) {
      if (r == p) continue;
      const float fr = M[2*(r*n+p)], fi = M[2*(r*n+p)+1];
      for (int c = 0; c < n; ++c) {
        const float ar = M[2*(p*n+c)], ai = M[2*(p*n+c)+1];
        M[2*(r*n+c)]   -= fr*ar - fi*ai;
        M[2*(r*n+c)+1] -= fr*ai + fi*ar;
        const float br = I[2*(p*n+c)], bi = I[2*(p*n+c)+1];
        I[2*(r*n+c)]   -= fr*br - fi*bi;
        I[2*(r*n+c)+1] -= fr*bi + fi*br;
      }
    }
    __syncthreads();
  }
}

// E = G - T - T^H + I  (batched n x n)
__global__ void ebuild_kernel(const float* G, const float* T, float* E, int n) {
  const int b = blockIdx.x;
  const long long bs = 2LL * n * n;
  const float* Gb = G + b * bs; const float* Tb = T + b * bs; float* Eb = E + b * bs;
  for (int e = threadIdx.x; e < n * n; e += blockDim.x) {
    const int m = e / n, c = e % n;
    const int et = c * n + m;
    Eb[2*e]   = Gb[2*e]   - Tb[2*e]   - Tb[2*et]   + ((m == c) ? 1.f : 0.f);
    Eb[2*e+1] = Gb[2*e+1] - Tb[2*e+1] + Tb[2*et+1];
  }
}

// W = 0.5 * (Inv + Inv^H)  (batched)
__global__ void hermw_kernel(const float* inv, long long bstrideF, float* Wo, int n) {
  const int b = blockIdx.x;
  const float* Ib = inv + (long long)b * bstrideF;
  float* Wb = Wo + (long long)b * 2LL * n * n;
  for (int e = threadIdx.x; e < n * n; e += blockDim.x) {
    const int m = e / n, c = e % n;
    const int et = c * n + m;
    Wb[2*e]   = 0.5f * (Ib[2*e]   + Ib[2*et]);
    Wb[2*e+1] = 0.5f * (Ib[2*e+1] - Ib[2*et+1]);
  }
}

// Per-user P normalization + pack to out rows 0..15.
__global__ void normp_kernel(const float* Pk, float* out, float pt) {
  const int k = blockIdx.x, tid = threadIdx.x;
  __shared__ float red[64];
  const float* Pb = Pk + k * 256;
  float s = 0.f;
  for (int i = tid; i < 256; i += 64) { const float v = Pb[i]; s += v * v; }
  red[tid] = s; __syncthreads();
  for (int w = 32; w > 0; w >>= 1) { if (tid < w) red[tid] += red[tid + w]; __syncthreads(); }
  const float scale = sqrtf(pt) / sqrtf(red[0] + 1e-30f);
  for (int e = tid; e < 128; e += 64) {
    const int u = e >> 3, m = e & 7;
    const int pos = u * 256 + m * 32 + k;
    out[pos]         = Pb[2*e]   * scale;
    out[36864 + pos] = Pb[2*e+1] * scale;
  }
}

// Global F normalization + pack to out rows 16..143.
__global__ void normf_kernel(const float* Fl, float* out, float pmax) {
  __shared__ float red[256];
  const int tid = threadIdx.x;
  float s = 0.f;
  for (int i = tid; i < 65536; i += 256) { const float v = Fl[i]; s += v * v; }
  red[tid] = s; __syncthreads();
  for (int w = 128; w > 0; w >>= 1) { if (tid < w) red[tid] += red[tid + w]; __syncthreads(); }
  const float scale = sqrtf(pmax) / sqrtf(red[0] + 1e-30f);
  for (int e = tid; e < 32768; e += 256) {
    const int l = e >> 10, t = (e >> 3) & 127, m = e & 7;
    const int pos = (16 + t) * 256 + m * 32 + l;
    out[pos]         = Fl[2*e]   * scale;
    out[36864 + pos] = Fl[2*e+1] * scale;
  }
}

// ---------------------------------------------------------------------------

static inline COp PLop(const float* p, long long plane) { COp o; o.p = p; o.s = 1; o.o = plane; return o; }
static inline COp WSop(const float* p)                  { COp o; o.p = p; o.s = 2; o.o = 1;     return o; }

static inline GemmArgs GA0() {
  GemmArgs g{};
  g.alpha = 1.f; g.dscale = 1.f; g.nb0 = 1; g.nbatch = 1; g.nred = 1;
  g.Cs = 2; g.Co = 1;
  return g;
}

extern "C" void kernel_launch(void* const* d_in, const int* in_sizes, int n_in,
                              void* d_out, int out_size, void* d_ws, size_t ws_size,
                              hipStream_t stream) {
  (void)in_sizes; (void)n_in; (void)out_size; (void)ws_size;
  const float* P  = (const float*)d_in[0];   // (2,16,8,32)
  const float* F  = (const float*)d_in[1];   // (2,128,8,32)
  const float* Th = (const float*)d_in[2];   // (2,2048,2048)
  const float* HU = (const float*)d_in[3];   // (2,128,16,32)
  const float* HD = (const float*)d_in[4];   // (2,16,128,32)
  const float* VU = (const float*)d_in[5];   // (2,128,2048)
  const float* VD = (const float*)d_in[6];   // (2,2048,128)
  const float* Jm = (const float*)d_in[7];   // (2,16,16,32,32)
  const float* GU = (const float*)d_in[8];   // (2,2048,16,32)
  const float* GD = (const float*)d_in[9];   // (2,16,2048,32)
  const float* Z  = (const float*)d_in[10];  // (2,16,2048,32)
  const float* Ht = (const float*)d_in[11];  // (2,128,128)
  float* out = (float*)d_out;
  float* ws  = (float*)d_ws;

  long long off = 0;
  auto alloc = [&](long long cplx) { float* p = ws + off; off += 2 * cplx; return p; };

  float* T1   = alloc(128LL*2048);
  float* HUb  = alloc(128LL*512);
  float* T2   = alloc(2048LL*128);
  float* HDb  = alloc(16LL*128*32);
  float* ZTw  = alloc(32LL*16*2048);
  float* Jb   = alloc(16LL*16*32*32);
  float* Pck  = alloc(32LL*16*2048);   // packed (l,d,m) for Z / GD (reused)
  float* sF   = alloc(128LL*128);
  float* HP   = alloc(32LL*128*8);
  float* AU   = alloc(128LL*128);
  float* T3   = alloc(128LL*128);
  float* HF   = alloc(32LL*16*8);
  float* JP   = alloc(32LL*32*16*8);
  float* AD   = alloc(32LL*16*16);
  float* gAU  = alloc(2LL*128*128);
  float* gAD  = alloc(32LL*2*16*16);
  float* UU   = alloc(32LL*128*8);
  float* UD   = alloc(32LL*16*8);
  float* TU   = alloc(32LL*8*8);
  float* T5   = alloc(32LL*128*8);
  float* GE   = alloc(32LL*8*8);
  float* EU   = alloc(32LL*8*8);
  float* TD   = alloc(32LL*8*8);
  float* T5D  = alloc(32LL*16*8);
  float* GDE  = alloc(32LL*8*8);
  float* ED   = alloc(32LL*8*8);
  float* gEU  = alloc(32LL*2*8*8);
  float* gED  = alloc(32LL*2*8*8);
  float* WU   = alloc(32LL*8*8);
  float* WD   = alloc(32LL*8*8);
  float* T6   = alloc(32LL*128*8);
  float* SU   = alloc(128LL*128);
  float* T7   = alloc(32LL*16*8);
  float* SD   = alloc(32LL*16*16);
  float* T8   = alloc(32LL*128*16);
  float* BU   = alloc(32LL*16*16);
  float* T9   = alloc(32LL*32*16*16);
  float* T10  = alloc(32LL*16*128);
  float* BD   = alloc(128LL*128);
  float* T11  = alloc(128LL*128);
  float* T12  = alloc(32LL*16*8);
  float* RHSP = alloc(32LL*16*8);
  float* gBU  = alloc(32LL*2*16*16);
  float* Pko  = alloc(32LL*16*8);
  float* T13  = alloc(32LL*128*8);
  float* RHSF = alloc(32LL*128*8);
  float* gBD  = alloc(2LL*128*128);
  float* Flo  = alloc(32LL*128*8);

  auto run = [&](GemmArgs& g) {
    dim3 grid((g.M + 15) / 16, (g.N + 15) / 16, g.nbatch);
    zgemm_wmma<<<grid, dim3(32), 0, stream>>>(g);
  };

  // ---- big TDM-staged GEMMs (Kd=2048) --------------------------------------
  // 1) T1 = VU @ Theta
  { BigArgs a{}; a.Ap=VU; a.Ao=262144; a.Bp=Th; a.Bo=4194304; a.Cp=T1;
    a.M=128; a.N=2048; a.Kd=2048; a.rsA=2048; a.rsB=2048; a.rsC=2048; a.csC=1;
    zgemm_big<0,0><<<dim3(8,32,1),128,0,stream>>>(a); }
  // 2) HUbar = HU + T1 @ GU(2048x512)
  { BigArgs a{}; a.Ap=T1; a.Bp=GU; a.Bo=1048576; a.Cp=HUb;
    a.M=128; a.N=512; a.Kd=2048; a.rsA=2048; a.rsB=512; a.rsC=512; a.csC=1;
    a.useD=1; a.Dp=HU; a.Do=65536; a.rsD=512; a.csD=1;
    zgemm_big<1,0><<<dim3(8,8,1),128,0,stream>>>(a); }
  // 3) T2 = Theta @ VD
  { BigArgs a{}; a.Ap=Th; a.Ao=4194304; a.Bp=VD; a.Bo=262144; a.Cp=T2;
    a.M=2048; a.N=128; a.Kd=2048; a.rsA=2048; a.rsB=128; a.rsC=128; a.csC=1;
    zgemm_big<0,0><<<dim3(128,2,1),128,0,stream>>>(a); }
  // 4) HDbar = HD + GD_l @ T2 :  pack GD -> (l,d,m), then batched big GEMM
  packzl_kernel<<<dim3(4096),dim3(256),0,stream>>>(GD, Pck);
  { BigArgs a{}; a.Ap=Pck; a.Bp=T2; a.Cp=HDb;
    a.M=16; a.N=128; a.Kd=2048; a.rsA=2048; a.sA0=32768; a.rsB=128;
    a.rsC=4096; a.csC=32; a.sC0=1;
    a.useD=1; a.Dp=HD; a.Do=65536; a.rsD=4096; a.csD=32; a.sD0=1;
    zgemm_big<1,1><<<dim3(1,2,32),128,0,stream>>>(a); }
  // 5) ZT_l = Z_l @ Theta :  pack Z -> (l,d,m) (reuse Pck), batched big GEMM
  packzl_kernel<<<dim3(4096),dim3(256),0,stream>>>(Z, Pck);
  { BigArgs a{}; a.Ap=Pck; a.Bp=Th; a.Bo=4194304; a.Cp=ZTw;
    a.M=16; a.N=2048; a.Kd=2048; a.rsA=2048; a.sA0=32768; a.rsB=2048;
    a.rsC=2048; a.csC=1; a.sC0=32768;
    zgemm_big<1,0><<<dim3(1,32,32),128,0,stream>>>(a); }
  // 6) Jbar = J + ZT_l @ GU(2048x512)
  { BigArgs a{}; a.Ap=ZTw; a.Bp=GU; a.Bo=1048576; a.Cp=Jb;
    a.M=16; a.N=512; a.Kd=2048; a.rsA=2048; a.sA0=32768; a.rsB=512;
    a.rsC=16384; a.csC=32; a.sC0=1;
    a.useD=1; a.Dp=Jm; a.Do=262144; a.rsD=16384; a.csD=32; a.sD0=1;
    zgemm_big<1,0><<<dim3(1,8,32),128,0,stream>>>(a); }

  // ---- generic WMMA GEMMs --------------------------------------------------
  // 7) sumF = F F^H
  { GemmArgs g = GA0(); g.A=PLop(F,32768); g.B=PLop(F,32768); g.Cp=sF;
    g.M=128; g.N=128; g.Kd=256; g.rsA=256; g.csA=1; g.rsB=1; g.csB=256; g.conjB=1;
    g.rsC=128; g.csC=1; run(g); }
  // 8) HP_k = HUbar_k @ P_k
  { GemmArgs g = GA0(); g.A=WSop(HUb); g.B=PLop(P,4096); g.Cp=HP;
    g.M=128; g.N=8; g.Kd=16; g.rsA=512; g.csA=32; g.sA0=1; g.rsB=256; g.csB=32; g.sB0=1;
    g.rsC=8; g.csC=1; g.sC0=1024; g.nb0=32; g.nbatch=32; run(g); }
  // 9) AU = sum_k HP HP^H + sigma_u^2 I
  { GemmArgs g = GA0(); g.A=WSop(HP); g.B=WSop(HP); g.Cp=AU;
    g.M=128; g.N=128; g.Kd=8; g.rsA=8; g.csA=1; g.rsB=1; g.csB=8; g.conjB=1;
    g.nred=32; g.sAr=1024; g.sBr=1024; g.rsC=128; g.csC=1; g.diag=0.01f; run(g); }
  // 10) T3 = Htilde @ sumF ; 11) AU += T3 @ Htilde^H
  { GemmArgs g = GA0(); g.A=PLop(Ht,16384); g.B=WSop(sF); g.Cp=T3;
    g.M=128; g.N=128; g.Kd=128; g.rsA=128; g.csA=1; g.rsB=128; g.csB=1; g.rsC=128; g.csC=1; run(g); }
  { GemmArgs g = GA0(); g.A=WSop(T3); g.B=PLop(Ht,16384); g.Cp=AU;
    g.M=128; g.N=128; g.Kd=128; g.rsA=128; g.csA=1; g.rsB=1; g.csB=128; g.conjB=1;
    g.rsC=128; g.csC=1; g.beta=1; run(g); }
  // 12) HF_l = HDbar_l @ F_l
  { GemmArgs g = GA0(); g.A=WSop(HDb); g.B=PLop(F,32768); g.Cp=HF;
    g.M=16; g.N=8; g.Kd=128; g.rsA=4096; g.csA=32; g.sA0=1; g.rsB=256; g.csB=32; g.sB0=1;
    g.rsC=8; g.csC=1; g.sC0=128; g.nb0=32; g.nbatch=32; run(g); }
  // 13) JP_{k,l} = Jbar_{k,l} @ P_k
  { GemmArgs g = GA0(); g.A=WSop(Jb); g.B=PLop(P,4096); g.Cp=JP;
    g.M=16; g.N=8; g.Kd=16; g.rsA=16384; g.csA=1024; g.sA0=32; g.sA1=1;
    g.rsB=256; g.csB=32; g.sB0=1; g.sB1=0;
    g.rsC=8; g.csC=1; g.sC0=4096; g.sC1=128; g.nb0=32; g.nbatch=1024; run(g); }
  // 14) AD_l = HF HF^H + sigma_d^2 I ; 15) AD_l += sum_k JP JP^H
  { GemmArgs g = GA0(); g.A=WSop(HF); g.B=WSop(HF); g.Cp=AD;
    g.M=16; g.N=16; g.Kd=8; g.rsA=8; g.csA=1; g.sA0=128; g.rsB=1; g.csB=8; g.sB0=128; g.conjB=1;
    g.rsC=16; g.csC=1; g.sC0=256; g.diag=0.01f; g.nb0=32; g.nbatch=32; run(g); }
  { GemmArgs g = GA0(); g.A=WSop(JP); g.B=WSop(JP); g.Cp=AD;
    g.M=16; g.N=16; g.Kd=8; g.rsA=8; g.csA=1; g.sA0=128; g.nred=32; g.sAr=4096;
    g.rsB=1; g.csB=8; g.sB0=128; g.sBr=4096; g.conjB=1;
    g.rsC=16; g.csC=1; g.sC0=256; g.beta=1; g.nb0=32; g.nbatch=32; run(g); }
  // 16/17) invert AU, AD
  cinv_kernel<<<dim3(1),  dim3(128), 0, stream>>>((const float*)AU, 2LL, 1LL, 128LL, 1LL, 0LL,   gAU, 65536LL, 128);
  cinv_kernel<<<dim3(32), dim3(64),  0, stream>>>((const float*)AD, 2LL, 1LL, 16LL,  1LL, 256LL, gAD, 1024LL,  16);
  // 18) UU_k = AUinv @ HP_k ; 19) UD_l = ADinv_l @ HF_l
  { GemmArgs g = GA0(); g.A=WSop(gAU+32768); g.B=WSop(HP); g.Cp=UU;
    g.M=128; g.N=8; g.Kd=128; g.rsA=128; g.csA=1; g.rsB=8; g.csB=1; g.sB0=1024;
    g.rsC=8; g.csC=1; g.sC0=1024; g.nb0=32; g.nbatch=32; run(g); }
  { GemmArgs g = GA0(); g.A=WSop(gAD+512); g.B=WSop(HF); g.Cp=UD;
    g.M=16; g.N=8; g.Kd=16; g.rsA=16; g.csA=1; g.sA0=512; g.rsB=8; g.csB=1; g.sB0=128;
    g.rsC=8; g.csC=1; g.sC0=128; g.nb0=32; g.nbatch=32; run(g); }
  // 20-23) uplink E
  { GemmArgs g = GA0(); g.A=WSop(HP); g.B=WSop(UU); g.Cp=TU;
    g.M=8; g.N=8; g.Kd=128; g.rsA=1; g.csA=8; g.sA0=1024; g.conjA=1;
    g.rsB=8; g.csB=1; g.sB0=1024; g.rsC=8; g.csC=1; g.sC0=64; g.nb0=32; g.nbatch=32; run(g); }
  { GemmArgs g = GA0(); g.A=WSop(AU); g.B=WSop(UU); g.Cp=T5;
    g.M=128; g.N=8; g.Kd=128; g.rsA=128; g.csA=1; g.rsB=8; g.csB=1; g.sB0=1024;
    g.rsC=8; g.csC=1; g.sC0=1024; g.nb0=32; g.nbatch=32; run(g); }
  { GemmArgs g = GA0(); g.A=WSop(UU); g.B=WSop(T5); g.Cp=GE;
    g.M=8; g.N=8; g.Kd=128; g.rsA=1; g.csA=8; g.sA0=1024; g.conjA=1;
    g.rsB=8; g.csB=1; g.sB0=1024; g.rsC=8; g.csC=1; g.sC0=64; g.nb0=32; g.nbatch=32; run(g); }
  ebuild_kernel<<<dim3(32), dim3(64), 0, stream>>>((const float*)GE, (const float*)TU, EU, 8);
  // 24-27) downlink E
  { GemmArgs g = GA0(); g.A=WSop(HF); g.B=WSop(UD); g.Cp=TD;
    g.M=8; g.N=8; g.Kd=16; g.rsA=1; g.csA=8; g.sA0=128; g.conjA=1;
    g.rsB=8; g.csB=1; g.sB0=128; g.rsC=8; g.csC=1; g.sC0=64; g.nb0=32; g.nbatch=32; run(g); }
  { GemmArgs g = GA0(); g.A=WSop(AD); g.B=WSop(UD); g.Cp=T5D;
    g.M=16; g.N=8; g.Kd=16; g.rsA=16; g.csA=1; g.sA0=256; g.rsB=8; g.csB=1; g.sB0=128;
    g.rsC=8; g.csC=1; g.sC0=128; g.nb0=32; g.nbatch=32; run(g); }
  { GemmArgs g = GA0(); g.A=WSop(UD); g.B=WSop(T5D); g.Cp=GDE;
    g.M=8; g.N=8; g.Kd=16; g.rsA=1; g.csA=8; g.sA0=128; g.conjA=1;
    g.rsB=8; g.csB=1; g.sB0=128; g.rsC=8; g.csC=1; g.sC0=64; g.nb0=32; g.nbatch=32; run(g); }
  ebuild_kernel<<<dim3(32), dim3(64), 0, stream>>>((const float*)GDE, (const float*)TD, ED, 8);
  // 28/29) W = hermitize(inv(E))
  cinv_kernel<<<dim3(32), dim3(64), 0, stream>>>((const float*)EU, 2LL, 1LL, 8LL, 1LL, 64LL, gEU, 256LL, 8);
  cinv_kernel<<<dim3(32), dim3(64), 0, stream>>>((const float*)ED, 2LL, 1LL, 8LL, 1LL, 64LL, gED, 256LL, 8);
  hermw_kernel<<<dim3(32), dim3(64), 0, stream>>>((const float*)(gEU+128), 256LL, WU, 8);
  hermw_kernel<<<dim3(32), dim3(64), 0, stream>>>((const float*)(gED+128), 256LL, WD, 8);
  // 30/31) SU
  { GemmArgs g = GA0(); g.A=WSop(UU); g.B=WSop(WU); g.Cp=T6;
    g.M=128; g.N=8; g.Kd=8; g.rsA=8; g.csA=1; g.sA0=1024; g.rsB=8; g.csB=1; g.sB0=64;
    g.rsC=8; g.csC=1; g.sC0=1024; g.nb0=32; g.nbatch=32; run(g); }
  { GemmArgs g = GA0(); g.A=WSop(T6); g.B=WSop(UU); g.Cp=SU;
    g.M=128; g.N=128; g.Kd=8; g.rsA=8; g.csA=1; g.nred=32; g.sAr=1024;
    g.rsB=1; g.csB=8; g.sBr=1024; g.conjB=1; g.rsC=128; g.csC=1; run(g); }
  // 32/33) SD
  { GemmArgs g = GA0(); g.A=WSop(UD); g.B=WSop(WD); g.Cp=T7;
    g.M=16; g.N=8; g.Kd=8; g.rsA=8; g.csA=1; g.sA0=128; g.rsB=8; g.csB=1; g.sB0=64;
    g.rsC=8; g.csC=1; g.sC0=128; g.nb0=32; g.nbatch=32; run(g); }
  { GemmArgs g = GA0(); g.A=WSop(T7); g.B=WSop(UD); g.Cp=SD;
    g.M=16; g.N=16; g.Kd=8; g.rsA=8; g.csA=1; g.sA0=128; g.rsB=1; g.csB=8; g.sB0=128; g.conjB=1;
    g.rsC=16; g.csC=1; g.sC0=256; g.nb0=32; g.nbatch=32; run(g); }
  // 34/35) BU part 1
  { GemmArgs g = GA0(); g.A=WSop(SU); g.B=WSop(HUb); g.Cp=T8;
    g.M=128; g.N=16; g.Kd=128; g.rsA=128; g.csA=1; g.rsB=512; g.csB=32; g.sB0=1;
    g.rsC=16; g.csC=1; g.sC0=2048; g.nb0=32; g.nbatch=32; run(g); }
  { GemmArgs g = GA0(); g.A=WSop(HUb); g.B=WSop(T8); g.Cp=BU;
    g.M=16; g.N=16; g.Kd=128; g.rsA=32; g.csA=512; g.sA0=1; g.conjA=1;
    g.rsB=16; g.csB=1; g.sB0=2048; g.rsC=16; g.csC=1; g.sC0=256; g.diag=1.6f;
    g.nb0=32; g.nbatch=32; run(g); }
  // 36/37) BU part 2
  { GemmArgs g = GA0(); g.A=WSop(SD); g.B=WSop(Jb); g.Cp=T9;
    g.M=16; g.N=16; g.Kd=16; g.rsA=16; g.csA=1; g.sA0=0; g.sA1=256;
    g.rsB=16384; g.csB=1024; g.sB0=32; g.sB1=1;
    g.rsC=16; g.csC=1; g.sC0=8192; g.sC1=256; g.nb0=32; g.nbatch=1024; run(g); }
  { GemmArgs g = GA0(); g.A=WSop(Jb); g.B=WSop(T9); g.Cp=BU;
    g.M=16; g.N=16; g.Kd=16; g.rsA=1024; g.csA=16384; g.sA0=32; g.nred=32; g.sAr=1; g.conjA=1;
    g.rsB=16; g.csB=1; g.sB0=8192; g.sBr=256; g.rsC=16; g.csC=1; g.sC0=256; g.beta=1;
    g.nb0=32; g.nbatch=32; run(g); }
  // 38/39) BD part 1
  { GemmArgs g = GA0(); g.A=WSop(SD); g.B=WSop(HDb); g.Cp=T10;
    g.M=16; g.N=128; g.Kd=16; g.rsA=16; g.csA=1; g.sA0=256; g.rsB=4096; g.csB=32; g.sB0=1;
    g.rsC=128; g.csC=1; g.sC0=2048; g.nb0=32; g.nbatch=32; run(g); }
  { GemmArgs g = GA0(); g.A=WSop(HDb); g.B=WSop(T10); g.Cp=BD;
    g.M=128; g.N=128; g.Kd=16; g.rsA=32; g.csA=4096; g.nred=32; g.sAr=1; g.conjA=1;
    g.rsB=128; g.csB=1; g.sBr=2048; g.rsC=128; g.csC=1; g.diag=0.02f; run(g); }
  // 40/41) BD part 2
  { GemmArgs g = GA0(); g.A=WSop(SU); g.B=PLop(Ht,16384); g.Cp=T11;
    g.M=128; g.N=128; g.Kd=128; g.rsA=128; g.csA=1; g.rsB=128; g.csB=1; g.rsC=128; g.csC=1; run(g); }
  { GemmArgs g = GA0(); g.A=PLop(Ht,16384); g.B=WSop(T11); g.Cp=BD;
    g.M=128; g.N=128; g.Kd=128; g.rsA=1; g.csA=128; g.conjA=1;
    g.rsB=128; g.csB=1; g.rsC=128; g.csC=1; g.beta=1; run(g); }
  // 42/43) RHS for P
  { GemmArgs g = GA0(); g.A=WSop(HUb); g.B=WSop(UU); g.Cp=T12;
    g.M=16; g.N=8; g.Kd=128; g.rsA=32; g.csA=512; g.sA0=1; g.conjA=1;
    g.rsB=8; g.csB=1; g.sB0=1024; g.rsC=8; g.csC=1; g.sC0=128; g.nb0=32; g.nbatch=32; run(g); }
  { GemmArgs g = GA0(); g.A=WSop(T12); g.B=WSop(WU); g.Cp=RHSP;
    g.M=16; g.N=8; g.Kd=8; g.rsA=8; g.csA=1; g.sA0=128; g.rsB=8; g.csB=1; g.sB0=64;
    g.rsC=8; g.csC=1; g.sC0=128; g.nb0=32; g.nbatch=32; run(g); }
  // 44/45) P_k = BUinv_k RHSP_k
  cinv_kernel<<<dim3(32), dim3(64), 0, stream>>>((const float*)BU, 2LL, 1LL, 16LL, 1LL, 256LL, gBU, 1024LL, 16);
  { GemmArgs g = GA0(); g.A=WSop(gBU+512); g.B=WSop(RHSP); g.Cp=Pko;
    g.M=16; g.N=8; g.Kd=16; g.rsA=16; g.csA=1; g.sA0=512; g.rsB=8; g.csB=1; g.sB0=128;
    g.rsC=8; g.csC=1; g.sC0=128; g.nb0=32; g.nbatch=32; run(g); }
  // 46/47) RHS for F
  { GemmArgs g = GA0(); g.A=WSop(HDb); g.B=WSop(UD); g.Cp=T13;
    g.M=128; g.N=8; g.Kd=16; g.rsA=32; g.csA=4096; g.sA0=1; g.conjA=1;
    g.rsB=8; g.csB=1; g.sB0=128; g.rsC=8; g.csC=1; g.sC0=1024; g.nb0=32; g.nbatch=32; run(g); }
  { GemmArgs g = GA0(); g.A=WSop(T13); g.B=WSop(WD); g.Cp=RHSF;
    g.M=128; g.N=8; g.Kd=8; g.rsA=8; g.csA=1; g.sA0=1024; g.rsB=8; g.csB=1; g.sB0=64;
    g.rsC=8; g.csC=1; g.sC0=1024; g.nb0=32; g.nbatch=32; run(g); }
  // 48/49) F_l = BDinv RHSF_l
  cinv_kernel<<<dim3(1), dim3(128), 0, stream>>>((const float*)BD, 2LL, 1LL, 128LL, 1LL, 0LL, gBD, 65536LL, 128);
  { GemmArgs g = GA0(); g.A=WSop(gBD+32768); g.B=WSop(RHSF); g.Cp=Flo;
    g.M=128; g.N=8; g.Kd=128; g.rsA=128; g.csA=1; g.rsB=8; g.csB=1; g.sB0=1024;
    g.rsC=8; g.csC=1; g.sC0=1024; g.nb0=32; g.nbatch=32; run(g); }
  // 50) normalize + pack
  normp_kernel<<<dim3(32), dim3(64),  0, stream>>>((const float*)Pko, out, 10.f);
  normf_kernel<<<dim3(1),  dim3(256), 0, stream>>>((const float*)Flo, out, 100.f);
}